// MultiHeadAttentionEdges_20117626814578
// MI455X (gfx1250) — compile-verified
//
#include <hip/hip_runtime.h>
#include <hip/hip_bf16.h>
#include <stdint.h>

typedef __attribute__((ext_vector_type(2))) float v2f;
typedef __attribute__((ext_vector_type(8))) float v8f;

__device__ __forceinline__ v8f wmma_f32_k4(v2f a, v2f b, v8f c) {
    // D = A(16x4 f32) * B(4x16 f32) + C(16x16 f32)
    return __builtin_amdgcn_wmma_f32_16x16x4_f32(
        /*neg_a=*/false, a, /*neg_b=*/false, b,
        /*c_mod=*/(short)0, c, /*reuse_a=*/false, /*reuse_b=*/false);
}

// Async DMA of 16 bytes global -> LDS, tracked by ASYNCcnt (CDNA5).
// GVS addressing: mem = SGPR base + 32-bit per-lane voffset.
__device__ __forceinline__ void async_copy_b128(uint32_t lds_byte_addr,
                                                const void* base,
                                                uint32_t voffset_bytes) {
    asm volatile("global_load_async_to_lds_b128 %0, %1, %2"
                 :
                 : "v"(lds_byte_addr), "v"(voffset_bytes), "s"(base)
                 : "memory");
}

__device__ __forceinline__ void wait_asynccnt0() {
    asm volatile("s_wait_asynccnt 0x0" ::: "memory");
}

// ---------------------------------------------------------------------------
// Generic one-wave 16x16-tile GEMM with bias: C[M,N] = A[M,K] @ B[K,N] + bias
// grid.x = M/16 row tiles, grid.y = N/16 col tiles, block = 32 (one wave)
// ---------------------------------------------------------------------------
__global__ void __launch_bounds__(32)
gemm16_bias(const float* __restrict__ A, const float* __restrict__ B,
            const float* __restrict__ bias, float* __restrict__ C,
            int K, int lda, int ldb, int ldc) {
    const int lane = threadIdx.x & 31;
    const int half = lane >> 4;     // 0: lanes 0-15, 1: lanes 16-31
    const int nl   = lane & 15;
    const int row0 = blockIdx.x * 16;
    const int col  = blockIdx.y * 16 + nl;
    const float* Ap = A + (size_t)(row0 + nl) * lda;   // A row = nl for this lane
    v8f c = {};
    for (int k = 0; k < K; k += 4) {
        const int ka = k + 2 * half;                   // lane-half selects K pair
        v2f a; a.x = Ap[ka];                a.y = Ap[ka + 1];
        v2f b; b.x = B[(size_t)ka * ldb + col]; b.y = B[(size_t)(ka + 1) * ldb + col];
        c = wmma_f32_k4(a, b, c);
    }
    const float bv = bias[col];
#pragma unroll
    for (int r = 0; r < 8; ++r) {
        const int m = r + 8 * half;                    // C layout: M = r + 8*half
        C[(size_t)(row0 + m) * ldc + col] = c[r] + bv;
    }
}

// ---------------------------------------------------------------------------
// Fused edge pipeline over 16 consecutive rows of flattened (b,i,j):
//   Ee = e@We+be ; inter = Ee*q*k*0.25 ; scores_pre = mask + head-sums(inter)
//   e_out = inter@Woe+boe
// block = 256 threads (8 waves); wave w owns output cols [16w, 16w+16)
// ---------------------------------------------------------------------------
__global__ void __launch_bounds__(256)
fused_edge(const float* __restrict__ e,
           const float* __restrict__ Qx, const float* __restrict__ Kx,
           const float* __restrict__ We, const float* __restrict__ be,
           const float* __restrict__ Woe, const float* __restrict__ boe,
           const float* __restrict__ mask,
           float* __restrict__ e_out, float* __restrict__ scores_pre) {
    __shared__ __align__(16) float sE[16 * 128];   // e tile
    __shared__ __align__(16) float sI[16 * 128];   // inter tile
    const int tid  = threadIdx.x;
    const int w    = tid >> 5;
    const int lane = tid & 31;
    const int half = lane >> 4;
    const int nl   = lane & 15;

    const size_t row0  = (size_t)blockIdx.x * 16;      // flattened (b,i,j) row
    const int    jbase = (int)(row0 & 511);
    const int    i     = (int)((row0 >> 9) & 511);
    const int    b     = (int)(row0 >> 18);

    // async DMA the 16x128 e tile into LDS (512 x b128; 2 per thread),
    // bypassing VGPRs entirely (ASYNCcnt-tracked)
    {
        const void* base = (const void*)(e + row0 * 128);
        const uint32_t lds0 = (uint32_t)(uintptr_t)&sE[0];
        async_copy_b128(lds0 + tid * 16u,           base, tid * 16u);
        async_copy_b128(lds0 + (tid + 256u) * 16u,  base, (tid + 256u) * 16u);
        wait_asynccnt0();
    }
    __syncthreads();

    const int col = w * 16 + nl;

    // GEMM1: Ee tile slice, K=128 (We slice stays hot in L2 across all blocks)
    v8f c = {};
    for (int k = 0; k < 128; k += 4) {
        const int ka = k + 2 * half;
        v2f a;  a.x = sE[nl * 128 + ka];        a.y = sE[nl * 128 + ka + 1];
        v2f bb; bb.x = We[ka * 128 + col];      bb.y = We[(ka + 1) * 128 + col];
        c = wmma_f32_k4(a, bb, c);
    }

    // elementwise: inter = (Ee + be) * Q[b,i,col] * K[b,j,col] / sqrt(dh=16)
    const float bev = be[col];
    const float qv  = Qx[((size_t)b * 512 + i) * 128 + col];
#pragma unroll
    for (int r = 0; r < 8; ++r) {
        const int m = r + 8 * half;
        const float kv = Kx[((size_t)b * 512 + jbase + m) * 128 + col];
        sI[m * 128 + col] = (c[r] + bev) * qv * kv * 0.25f;
    }
    __syncthreads();

    // pre-softmax scores: one (row m, head hh) per thread, sum over dh=16
    if (tid < 128) {
        const int m = tid >> 3, hh = tid & 7;
        float s = 0.f;
#pragma unroll
        for (int d = 0; d < 16; ++d) s += sI[m * 128 + hh * 16 + d];
        s += mask[((size_t)b * 512 + i) * 512 + jbase + m];
        scores_pre[(((size_t)b * 8 + hh) * 512 + i) * 512 + jbase + m] = s;
    }

    // GEMM2: e_out = inter @ Woe + boe (A from LDS, full K=128)
    v8f c2 = {};
    for (int k = 0; k < 128; k += 4) {
        const int ka = k + 2 * half;
        v2f a;  a.x = sI[nl * 128 + ka];        a.y = sI[nl * 128 + ka + 1];
        v2f bb; bb.x = Woe[ka * 128 + col];     bb.y = Woe[(ka + 1) * 128 + col];
        c2 = wmma_f32_k4(a, bb, c2);
    }
    const float bo = boe[col];
#pragma unroll
    for (int r = 0; r < 8; ++r) {
        const int m = r + 8 * half;
        e_out[(row0 + m) * 128 + col] = c2[r] + bo;
    }
}

// ---------------------------------------------------------------------------
// In-place softmax over rows of 512 (scores region of d_out)
// ---------------------------------------------------------------------------
__global__ void __launch_bounds__(256)
softmax512(float* __restrict__ s) {
    float* p = s + (size_t)blockIdx.x * 512;
    const int t = threadIdx.x;
    __shared__ float red[8];
    float a = p[t], b = p[t + 256];
    float m = fmaxf(a, b);
    for (int o = 16; o > 0; o >>= 1) m = fmaxf(m, __shfl_xor(m, o));
    if ((t & 31) == 0) red[t >> 5] = m;
    __syncthreads();
    float mg = red[0];
#pragma unroll
    for (int q = 1; q < 8; ++q) mg = fmaxf(mg, red[q]);
    __syncthreads();
    const float ea = __expf(a - mg), eb = __expf(b - mg);
    float sum = ea + eb;
    for (int o = 16; o > 0; o >>= 1) sum += __shfl_xor(sum, o);
    if ((t & 31) == 0) red[t >> 5] = sum;
    __syncthreads();
    float tot = 0.f;
#pragma unroll
    for (int q = 0; q < 8; ++q) tot += red[q];
    const float inv = 1.f / tot;
    p[t] = ea * inv;
    p[t + 256] = eb * inv;
}

// ---------------------------------------------------------------------------
// att[b,i,h*16+d] = sum_j scores[b,h,i,j] * Kx[b,j,h*16+d]   (K=512, WMMA)
// grid = (i-tiles=32, h=8, b=2), one wave per block
// ---------------------------------------------------------------------------
__global__ void __launch_bounds__(32)
att_kernel(const float* __restrict__ scores, const float* __restrict__ Kx,
           float* __restrict__ att) {
    const int it = blockIdx.x, h = blockIdx.y, b = blockIdx.z;
    const int lane = threadIdx.x & 31;
    const int half = lane >> 4, nl = lane & 15;
    const float* S  = scores + (((size_t)(b * 8 + h)) * 512 + it * 16) * 512;
    const float* Kp = Kx + (size_t)b * 512 * 128 + h * 16;
    v8f c = {};
    for (int k = 0; k < 512; k += 4) {
        const int ka = k + 2 * half;
        v2f a;  a.x = S[(size_t)nl * 512 + ka];  a.y = S[(size_t)nl * 512 + ka + 1];
        v2f bb; bb.x = Kp[(size_t)ka * 128 + nl]; bb.y = Kp[(size_t)(ka + 1) * 128 + nl];
        c = wmma_f32_k4(a, bb, c);
    }
#pragma unroll
    for (int r = 0; r < 8; ++r) {
        const int m = r + 8 * half;
        att[((size_t)b * 512 + it * 16 + m) * 128 + h * 16 + nl] = c[r];
    }
}

// ---------------------------------------------------------------------------
extern "C" void kernel_launch(void* const* d_in, const int* in_sizes, int n_in,
                              void* d_out, int out_size, void* d_ws, size_t ws_size,
                              hipStream_t stream) {
    const float* x    = (const float*)d_in[0];
    const float* e    = (const float*)d_in[1];
    const float* mask = (const float*)d_in[2];
    const float* Wq   = (const float*)d_in[3];
    const float* bq   = (const float*)d_in[4];
    const float* Wk   = (const float*)d_in[5];
    const float* bk   = (const float*)d_in[6];
    const float* We   = (const float*)d_in[7];
    const float* be   = (const float*)d_in[8];
    const float* Woh  = (const float*)d_in[9];
    const float* boh  = (const float*)d_in[10];
    const float* Woe  = (const float*)d_in[11];
    const float* boe  = (const float*)d_in[12];

    float* out    = (float*)d_out;
    float* x_out  = out;                                  // [2,512,128]
    float* e_out  = out + 131072;                         // [2,512,512,128]
    float* scores = out + 131072 + 67108864;              // [2,8,512,512]

    float* ws  = (float*)d_ws;
    float* Qx  = ws;                // [2,512,128]
    float* Kx  = ws + 131072;       // [2,512,128]
    float* att = ws + 262144;       // [2,512,128]

    const dim3 g64x8(64, 8);
    // Qx, Kx projections (M=1024, K=N=128)
    gemm16_bias<<<g64x8, 32, 0, stream>>>(x, Wq, bq, Qx, 128, 128, 128, 128);
    gemm16_bias<<<g64x8, 32, 0, stream>>>(x, Wk, bk, Kx, 128, 128, 128, 128);
    // fused Ee / inter / pre-softmax scores / e_out  (524288 rows / 16 per WG)
    fused_edge<<<32768, 256, 0, stream>>>(e, Qx, Kx, We, be, Woe, boe, mask,
                                          e_out, scores);
    // softmax over j (8192 rows of 512), in place
    softmax512<<<8192, 256, 0, stream>>>(scores);
    // att = scores @ Vx (Vx == Kx)
    att_kernel<<<dim3(32, 8, 2), 32, 0, stream>>>(scores, Kx, att);
    // x_out = att @ Woh + boh
    gemm16_bias<<<g64x8, 32, 0, stream>>>(att, Woh, boh, x_out, 128, 128, 128, 128);
}